// MyMarginLoss_63831803953310
// MI455X (gfx1250) — compile-verified
//
#include <hip/hip_runtime.h>

// MyMarginLoss on MI455X (gfx1250, wave32).
// Memory-bound: ~168 MB traffic -> ~7.2 us floor at 23.3 TB/s.
// Strategy: 1 wave = 16 rows. Two independent V_WMMA_F32_16X16X4_F32 chains:
//   acc  = masked row sums      (A = src * maskf, B = ones)
//   acc2 = row mask popcounts   (A = maskf,       B = ones)  -- exact in f32
// Masking is done by multiply (bools are 0/1 bytes -> v_cvt_f32_ubyte + mul),
// so the inner loop is loads + 2 cvt + 2 mul + 2 wmma per 4 columns.
// Deterministic two-stage reduction (no float atomics).

typedef __attribute__((ext_vector_type(2))) float v2f;
typedef __attribute__((ext_vector_type(8))) float v8f;

#define B_DIM 256
#define M_DIM 256
#define C_DIM 512
#define ROWS  (B_DIM * M_DIM)          // 65536
#define WAVES_PER_BLOCK 8
#define ROWS_PER_WAVE   16
#define ROWS_PER_BLOCK  (WAVES_PER_BLOCK * ROWS_PER_WAVE)  // 128
#define NUM_BLOCKS      (ROWS / ROWS_PER_BLOCK)            // 512

__global__ __launch_bounds__(256) void margin_partial_kernel(
    const float* __restrict__ src,             // [B,M,C] f32
    const int* __restrict__ target,            // [B,M] i32
    const unsigned char* __restrict__ mask_m,  // [B,M] bool(1B)
    const unsigned char* __restrict__ mask_c,  // [B,M,C] bool(1B)
    float* __restrict__ partial)               // [NUM_BLOCKS]
{
    const int lane = threadIdx.x & 31;
    const int wave = threadIdx.x >> 5;
    const long rowBase = (long)(blockIdx.x * WAVES_PER_BLOCK + wave) * ROWS_PER_WAVE;

    // A-matrix fragment addressing (32-bit A 16x4 layout):
    //   lanes 0-15: row = lane, K = {0,1}; lanes 16-31: row = lane-16, K = {2,3}
    const int r16 = lane & 15;
    const int kp  = (lane >> 4) * 2;   // 0 or 2
    const float*         rowPtr = src    + (rowBase + r16) * C_DIM + kp;
    const unsigned char* mPtr   = mask_c + (rowBase + r16) * C_DIM + kp;

    v8f acc  = {};                // 16x16 f32: row sums (replicated over N)
    v8f acc2 = {};                // 16x16 f32: row popcounts (replicated over N)
    v2f ones; ones.x = 1.0f; ones.y = 1.0f;   // B = all ones (layout-invariant)

    // 512 columns, 4 K-slices per WMMA pair -> 128 iterations, fully converged.
    for (int c = 0; c < C_DIM; c += 4) {
        v2f a = *(const v2f*)(rowPtr + c);                       // global_load_b64
        const unsigned int mw = *(const unsigned short*)(mPtr + c);
        v2f m;
        m.x = (float)(mw & 0xFFu);     // v_cvt_f32_ubyte0 (bools are 0/1)
        m.y = (float)(mw >> 8);        // v_cvt_f32_ubyte1
        a.x *= m.x;
        a.y *= m.y;
        // D = A(16x4) x ones(4x16) + C : two independent accumulation chains
        acc = __builtin_amdgcn_wmma_f32_16x16x4_f32(
            false, a, false, ones, (short)0, acc, false, false);
        acc2 = __builtin_amdgcn_wmma_f32_16x16x4_f32(
            false, m, false, ones, (short)0, acc2, false, false);
    }

    // Row extraction from D layout (sums AND counts share it):
    //   lanes 0-7  own rows 0-7   (D[r][n=lane] lives in VGPR r of lanes 0-15)
    //   lanes 16-23 own rows 8-15 (D[8+j][n=j] lives in VGPR j of lane 16+j)
    const int  sub       = lane & 7;
    const bool activeRow = ((lane & 15) < 8);
    const int  row       = (lane < 16) ? sub : (8 + sub);
    float s  = 0.0f;
    float sz = 0.0f;
#pragma unroll
    for (int v = 0; v < 8; ++v) {
        s  = (sub == v) ? acc[v]  : s;
        sz = (sub == v) ? acc2[v] : sz;
    }

    float per = 0.0f;
    if (activeRow) {
        const long gRow = rowBase + row;
        const int  tgt  = target[gRow];
        // Prefix-mask guarantee: compacted index == slot index, target < size.
        const float tmpt = src[gRow * C_DIM + tgt];
        const float val  = (sz - 1.0f - sz * tmpt + s) / fmaxf(sz, 1.0f);
        per = (mask_m[gRow] != 0 && sz > 0.5f) ? val : 0.0f;
    }

    // Deterministic wave reduction (inactive lanes contribute 0).
    per += __shfl_xor(per, 16, 32);
    per += __shfl_xor(per,  8, 32);
    per += __shfl_xor(per,  4, 32);
    per += __shfl_xor(per,  2, 32);
    per += __shfl_xor(per,  1, 32);

    __shared__ float lds[WAVES_PER_BLOCK];
    if (lane == 0) lds[wave] = per;
    __syncthreads();
    if (threadIdx.x == 0) {
        float t = 0.0f;
#pragma unroll
        for (int i = 0; i < WAVES_PER_BLOCK; ++i) t += lds[i];
        partial[blockIdx.x] = t;
    }
}

__global__ __launch_bounds__(256) void reduce_partials_kernel(
    const float* __restrict__ p, float* __restrict__ out, int n)
{
    float s = 0.0f;
    for (int i = threadIdx.x; i < n; i += 256) s += p[i];
    s += __shfl_xor(s, 16, 32);
    s += __shfl_xor(s,  8, 32);
    s += __shfl_xor(s,  4, 32);
    s += __shfl_xor(s,  2, 32);
    s += __shfl_xor(s,  1, 32);
    __shared__ float lds[8];
    const int wave = threadIdx.x >> 5;
    if ((threadIdx.x & 31) == 0) lds[wave] = s;
    __syncthreads();
    if (threadIdx.x == 0) {
        float t = 0.0f;
#pragma unroll
        for (int i = 0; i < 8; ++i) t += lds[i];
        out[0] = t;     // written every call -> deterministic, no init needed
    }
}

extern "C" void kernel_launch(void* const* d_in, const int* in_sizes, int n_in,
                              void* d_out, int out_size, void* d_ws, size_t ws_size,
                              hipStream_t stream) {
    // setup_inputs() order: source(f32), target(i32), mask_mentions(bool),
    // mask_candidates(bool). Output: single f32 scalar.
    const float*         src    = (const float*)d_in[0];
    const int*           target = (const int*)d_in[1];
    const unsigned char* mask_m = (const unsigned char*)d_in[2];
    const unsigned char* mask_c = (const unsigned char*)d_in[3];
    float* partial = (float*)d_ws;          // NUM_BLOCKS * 4 bytes of scratch

    margin_partial_kernel<<<NUM_BLOCKS, 256, 0, stream>>>(
        src, target, mask_m, mask_c, partial);
    reduce_partials_kernel<<<1, 256, 0, stream>>>(
        partial, (float*)d_out, NUM_BLOCKS);
}